// GraphConvertor_22342419874140
// MI455X (gfx1250) — compile-verified
//
#include <hip/hip_runtime.h>

// ---------------------------------------------------------------------------
// Problem constants (from the reference):
//   B=4, C=32, HW=256, PATCH=4  ->  pooled grid 64x64, N=4096 nodes
//   nodes: (4, 32, 4096)  = 524288 floats  (32 replicas of s[b,p] per batch)
//   adj  : (4, 4096, 4096) = 67108864 floats (sparse: <=4 nz per column)
// ---------------------------------------------------------------------------

#define NB      4
#define NC      32
#define HWF     256
#define PG      64            // pooled grid side
#define NNODES  4096          // PG*PG
#define NODES_PER_BATCH (NC * NNODES)    // 131072
#define NODES_TOTAL     (NB * NODES_PER_BATCH)
#define EPSV    1e-6f

typedef __attribute__((ext_vector_type(2))) float v2f;
typedef __attribute__((ext_vector_type(4))) float v4f;
typedef __attribute__((ext_vector_type(8))) float v8f;

// ---------------------------------------------------------------------------
// Kernel 1: streaming zero-fill of adj with 128-bit non-temporal stores.
// adj is 256MB (> 192MB L2), so NT avoids polluting L2 before the scatter.
// ---------------------------------------------------------------------------
__global__ __launch_bounds__(256) void zero_adj_kernel(v4f* __restrict__ adj, long n4) {
    long i      = (long)blockIdx.x * blockDim.x + threadIdx.x;
    long stride = (long)gridDim.x * blockDim.x;
    v4f z = {0.0f, 0.0f, 0.0f, 0.0f};
    for (; i < n4; i += stride) {
        __builtin_nontemporal_store(z, adj + i);
    }
}

// ---------------------------------------------------------------------------
// Kernel 2: channel-sum of 4x4 avg-pooled x_feat via V_WMMA_F32_16X16X4_F32.
//
// s[b,p] = (1/16) * sum_{c,dy,dx} x_feat[b, c, 4*pr+dy, 4*pc+dx]
//
// Mapped as S = A x ones:  D[m][n] = sum_k A[m][k].
// One wave handles 16 patches (fixed b, pr, group of 16 pc).
// A layout for 16x4 f32 (ISA 7.12.2): lane m (0-15) holds K=0,1 in v0,v1;
// lane m+16 holds K=2,3. With K = 4 consecutive dx pixels of one (c,dy) row:
//   lane = m + 16*half  loads float2 at x_feat[b, c, 4*pr+dy, 4*pc + 2*half].
// B = all-ones (layout-independent). Chain 128 WMMAs through C.
// D layout: acc[r] on lanes 0-15 = s[m=r], on lanes 16-31 = s[m=r+8].
// ---------------------------------------------------------------------------
__global__ __launch_bounds__(32) void pool_nodes_wmma_kernel(const float* __restrict__ xfeat,
                                                             float* __restrict__ nodes) {
    const int wave = blockIdx.x;        // 0..1023
    const int b    = wave >> 8;         // /256
    const int rem  = wave & 255;
    const int pr   = rem >> 2;          // 0..63 pooled row
    const int pg   = rem & 3;           // group of 16 pooled cols
    const int lane = threadIdx.x;
    const int m    = lane & 15;         // patch within group
    const int half = lane >> 4;         // 0 -> K=0,1 ; 1 -> K=2,3

    const int pc = pg * 16 + m;
    // x_feat[b, 0, 4*pr, 4*pc + 2*half]
    const float* base = xfeat
        + ((size_t)b * NC) * (HWF * HWF)
        + (size_t)(4 * pr) * HWF
        + (size_t)(4 * pc + 2 * half);

    v2f ones2; ones2[0] = 1.0f; ones2[1] = 1.0f;
    v8f acc = {};

    for (int c = 0; c < NC; ++c) {
        const float* cb = base + (size_t)c * (HWF * HWF);
#pragma unroll
        for (int dy = 0; dy < 4; ++dy) {
            v2f a = *(const v2f*)(cb + dy * HWF);
            // D = A x ones + C  : accumulates 4 K-pixels into 16 row sums
            acc = __builtin_amdgcn_wmma_f32_16x16x4_f32(
                /*neg_a=*/false, a, /*neg_b=*/false, ones2,
                /*c_mod=*/(short)0, acc, /*reuse_a=*/false, /*reuse_b=*/false);
        }
    }

    // Fan-out: nodes[b*131072 + k*4096 + p] = s[b,p] for k = 0..31.
    // Lanes 0-15 hold patches m=0..7 (acc[r] = s[r]), lanes 16-31 hold m=8..15.
    // Each of the 16 lanes sharing a value writes replicas k = m and k = m+16.
    const int p_base = pr * PG + pg * 16 + 8 * half;
    const size_t nb  = (size_t)b * NODES_PER_BATCH;
    const int k0     = m;
#pragma unroll
    for (int r = 0; r < 8; ++r) {
        float s = acc[r] * 0.0625f;     // avg-pool 1/16
        int p = p_base + r;
        nodes[nb + (size_t)k0        * NNODES + p] = s;
        nodes[nb + (size_t)(k0 + 16) * NNODES + p] = s;
    }
}

// ---------------------------------------------------------------------------
// Kernel 3: xv = 1 - avgpool(x_var) into LDS, then sparse scatter into adj.
// One block per batch. All (from,to) pairs are distinct -> plain stores.
// ---------------------------------------------------------------------------
__global__ __launch_bounds__(256) void var_adj_kernel(const float* __restrict__ xvar,
                                                      float* __restrict__ adj) {
    __shared__ float xv[NNODES];        // 16KB of the 320KB LDS
    const int b   = blockIdx.x;
    const int tid = threadIdx.x;

    const float* vb = xvar + (size_t)b * (HWF * HWF);
    for (int i = tid; i < NNODES; i += 256) {
        int pr = i >> 6, pc = i & 63;
        const float* rp = vb + (size_t)(4 * pr) * HWF + (size_t)(4 * pc);
        float sum = 0.0f;
#pragma unroll
        for (int dy = 0; dy < 4; ++dy) {
            v4f v = *(const v4f*)(rp + dy * HWF);
            sum += v[0] + v[1] + v[2] + v[3];
        }
        xv[i] = 1.0f - sum * 0.0625f;
    }
    __syncthreads();

    float* ab = adj + (size_t)b * NNODES * NNODES;
    const int dr[4] = { 1, -1, 0,  0 };
    const int dc[4] = { 0,  0, 1, -1 };
    for (int i = tid; i < NNODES; i += 256) {
        int r = i >> 6, c = i & 63;
        float self = xv[i];
#pragma unroll
        for (int d = 0; d < 4; ++d) {
            int nr = r + dr[d], nc = c + dc[d];
            if (nr >= 0 && nr < PG && nc >= 0 && nc < PG) {
                int from = nr * PG + nc;
                float diff = xv[from] - self;
                if (diff > EPSV) {
                    ab[(size_t)from * NNODES + i] = diff;   // adj[b, from, to=i]
                }
            }
        }
    }
}

// ---------------------------------------------------------------------------
// Host launch: zero-fill -> (WMMA pooling || var/adj scatter), stream-ordered.
// ---------------------------------------------------------------------------
extern "C" void kernel_launch(void* const* d_in, const int* in_sizes, int n_in,
                              void* d_out, int out_size, void* d_ws, size_t ws_size,
                              hipStream_t stream) {
    const float* x_feat = (const float*)d_in[0];
    const float* x_var  = (const float*)d_in[1];

    float* nodes = (float*)d_out;                 // 524288 floats
    float* adj   = nodes + NODES_TOTAL;           // 67108864 floats

    const long adj_f4 = (long)NB * NNODES * NNODES / 4;   // 16,777,216 x float4
    zero_adj_kernel<<<4096, 256, 0, stream>>>((v4f*)adj, adj_f4);

    pool_nodes_wmma_kernel<<<NB * PG * 4, 32, 0, stream>>>(x_feat, nodes);

    var_adj_kernel<<<NB, 256, 0, stream>>>(x_var, adj);
}